// GCNLayer_34016140984773
// MI455X (gfx1250) — compile-verified
//
#include <hip/hip_runtime.h>
#include <hip/hip_bf16.h>

typedef __attribute__((ext_vector_type(2))) float v2f;
typedef __attribute__((ext_vector_type(8))) float v8f;

#define D_FEAT 128

// ---------------------------------------------------------------------------
// Zero workspace (agg + deg), grid-stride.
// ---------------------------------------------------------------------------
__global__ void gcn_zero_kernel(float* __restrict__ p, size_t n) {
    size_t i = (size_t)blockIdx.x * blockDim.x + threadIdx.x;
    size_t stride = (size_t)gridDim.x * blockDim.x;
    for (; i < n; i += stride) p[i] = 0.0f;
}

// ---------------------------------------------------------------------------
// In-degree: deg[dst[e]] += 1.0f  (float atomics; exact up to 2^24 counts)
// ---------------------------------------------------------------------------
__global__ void gcn_degree_kernel(const int* __restrict__ dst,
                                  float* __restrict__ deg, int n_edges) {
    int e = blockIdx.x * blockDim.x + threadIdx.x;
    if (e < n_edges) {
        __hip_atomic_fetch_add(&deg[dst[e]], 1.0f,
                               __ATOMIC_RELAXED, __HIP_MEMORY_SCOPE_AGENT);
    }
}

// ---------------------------------------------------------------------------
// Fused norm + gather + scatter: one wave32 per edge.
//   agg[dst[e], :] += feat[src[e], :] * rsqrt(max(deg[src[e]],1))
// Each lane handles 4 of the 128 columns -> coalesced f32 atomics into L2.
// ---------------------------------------------------------------------------
__global__ void gcn_scatter_kernel(const float* __restrict__ feat,
                                   const float* __restrict__ deg,
                                   const int* __restrict__ src,
                                   const int* __restrict__ dst,
                                   float* __restrict__ agg, int n_edges) {
    int wave = (int)((blockIdx.x * (size_t)blockDim.x + threadIdx.x) >> 5);
    int lane = threadIdx.x & 31;
    if (wave >= n_edges) return;
    int s = src[wave];
    int d = dst[wave];
    float nrm = rsqrtf(fmaxf(deg[s], 1.0f));
    const float* fs = feat + (size_t)s * D_FEAT;
    float* ad = agg + (size_t)d * D_FEAT;
#pragma unroll
    for (int i = 0; i < 4; ++i) {
        int c = lane + i * 32;
        __hip_atomic_fetch_add(&ad[c], fs[c] * nrm,
                               __ATOMIC_RELAXED, __HIP_MEMORY_SCOPE_AGENT);
    }
}

// ---------------------------------------------------------------------------
// h = relu(agg @ W^T) + beta * feat    via V_WMMA_F32_16X16X4_F32.
// One wave per 16x16 output tile; K=128 swept in 32 steps of 4.
//
//   D[m,n] = sum_k agg[m,k] * W[n,k]
//   A fragment (16x4, MxK): lane l<16 -> M=l, K=0..1 in v[0..1];
//                           lane l>=16 -> M=l-16, K=2..3.
//   B fragment (4x16, KxN): mirrored with N in place of M (W row-major,
//                           so B[k,n] = W[n,k] -> contiguous float2 loads).
//   C/D (16x16 f32): VGPR r, lanes 0-15 -> M=r,N=lane; lanes 16-31 -> M=r+8.
// Grid sized exactly (no bounds check) so EXEC is all-ones for WMMA.
// ---------------------------------------------------------------------------
__global__ void gcn_gemm_kernel(const float* __restrict__ agg,
                                const float* __restrict__ W,
                                const float* __restrict__ feat,
                                const float* __restrict__ beta,
                                float* __restrict__ out,
                                int tiles_n /* col tiles = D/16 */) {
    int wave = (int)((blockIdx.x * (size_t)blockDim.x + threadIdx.x) >> 5);
    int lane = threadIdx.x & 31;
    int tile_m = wave / tiles_n;
    int tile_n = wave % tiles_n;
    int half = lane >> 4;     // 0: lanes 0-15, 1: lanes 16-31
    int l15  = lane & 15;

    const float* aptr = agg + (size_t)(tile_m * 16 + l15) * D_FEAT + half * 2;
    const float* bptr = W   + (size_t)(tile_n * 16 + l15) * D_FEAT + half * 2;

    v8f acc = {};
#pragma unroll 8
    for (int k = 0; k < D_FEAT; k += 4) {
        v2f a = *(const v2f*)(aptr + k);
        v2f b = *(const v2f*)(bptr + k);
        acc = __builtin_amdgcn_wmma_f32_16x16x4_f32(
            /*neg_a=*/false, a, /*neg_b=*/false, b,
            /*c_mod=*/(short)0, acc, /*reuse_a=*/false, /*reuse_b=*/false);
    }

    float bv = beta[0];
#pragma unroll
    for (int r = 0; r < 8; ++r) {
        int m = tile_m * 16 + r + half * 8;
        int n = tile_n * 16 + l15;
        size_t idx = (size_t)m * D_FEAT + n;
        float h = fmaxf(acc[r], 0.0f);
        out[idx] = h + bv * feat[idx];
    }
}

// ---------------------------------------------------------------------------
extern "C" void kernel_launch(void* const* d_in, const int* in_sizes, int n_in,
                              void* d_out, int out_size, void* d_ws, size_t ws_size,
                              hipStream_t stream) {
    const float* feat = (const float*)d_in[0];
    const float* W    = (const float*)d_in[1];
    const float* beta = (const float*)d_in[2];
    const int*   src  = (const int*)d_in[3];
    const int*   dst  = (const int*)d_in[4];
    float*       out  = (float*)d_out;

    const int D = D_FEAT;                 // in_sizes[1] == D*D
    const int N = in_sizes[0] / D;        // 100000
    const int n_edges = in_sizes[3];      // 1600000

    float* agg = (float*)d_ws;                       // N*D floats (51.2 MB)
    float* deg = agg + (size_t)N * D;                // N floats

    // 1) zero agg + deg
    size_t zn = (size_t)N * D + (size_t)N;
    gcn_zero_kernel<<<2048, 256, 0, stream>>>(agg, zn);

    // 2) in-degree
    gcn_degree_kernel<<<(n_edges + 255) / 256, 256, 0, stream>>>(dst, deg, n_edges);

    // 3) fused norm + gather + scatter (1 wave / edge, 8 waves / block)
    gcn_scatter_kernel<<<(n_edges + 7) / 8, 256, 0, stream>>>(
        feat, deg, src, dst, agg, n_edges);

    // 4) WMMA GEMM + fused relu/residual epilogue (1 wave / 16x16 tile)
    int tiles_m = N / 16;                  // 6250
    int tiles_n = D / 16;                  // 8
    int n_tiles = tiles_m * tiles_n;       // 50000, divisible by 8
    gcn_gemm_kernel<<<n_tiles / 8, 256, 0, stream>>>(agg, W, feat, beta, out, tiles_n);
}